// CategoricalActivation_18184891531354
// MI455X (gfx1250) — compile-verified
//
#include <hip/hip_runtime.h>
#include <hip/hip_bf16.h>
#include <stdint.h>

// Problem constants (fixed by the reference)
#define S_DIM 2048
#define B_DIM 16
#define H_DIM 1024
#define BH    (B_DIM * H_DIM)        // 16384, power of two
#define NC    5
#define S_TILE 16                    // s-rows per block in main kernel
#define BH_PER_BLOCK 1024            // 256 threads x f32x4

typedef unsigned int uint32x4 __attribute__((ext_vector_type(4)));
typedef int          int32x8  __attribute__((ext_vector_type(8)));
typedef int          int32x4  __attribute__((ext_vector_type(4)));
typedef float        f32x4    __attribute__((ext_vector_type(4)));

__device__ __forceinline__ float softsign_f(float v) {
    return v / (1.0f + __builtin_fabsf(v));
}

// ---------------------------------------------------------------------------
// TDM: 1-D contiguous f32 tile, global -> LDS.  Descriptor built per CDNA5
// CDNA5 ISA (group0 128b, group1 256b).
// 6-arg builtin form (this toolchain): (g0, g1, g2, g3, g4, cpol).
// ---------------------------------------------------------------------------
__device__ __forceinline__ void tdm_load_1d_f32(uint32_t lds_byte_off,
                                                const void* gptr,
                                                uint32_t nelem /* f32 count, <=65535 */) {
    uint64_t ga = (uint64_t)(uintptr_t)gptr;
    uint32x4 g0;
    g0[0] = 1u;                                   // count=1 valid, is_restore=0, no gather
    g0[1] = lds_byte_off;                         // lds_addr [63:32]
    g0[2] = (uint32_t)ga;                         // global_addr lo  [95:64]
    g0[3] = (uint32_t)((ga >> 32) & 0x01FFFFFFu)  // global_addr hi  [120:96]
          | (2u << 30);                           // type=2 ("image") [127:126]
    int32x8 g1;
    g1[0] = (int)(2u << 16);                      // workgroup_mask=0, data_size=2 (4B)
    g1[1] = (int)((nelem & 0xFFFFu) << 16);       // tensor_dim0 lo16 at [63:48]
    g1[2] = (int)((nelem >> 16) & 0xFFFFu)        // tensor_dim0 hi16 at [79:64]
          | (1 << 16);                            // tensor_dim1 lo16 = 1
    g1[3] = (int)(nelem << 16);                   // tensor_dim1 hi=0 | tile_dim0 at [127:112]
    g1[4] = 1;                                    // tile_dim1=1, tile_dim2=0
    g1[5] = (int)nelem;                           // tensor_dim0_stride lo32
    g1[6] = 0;                                    // stride0 hi16 | tensor_dim1_stride lo16
    g1[7] = 0;
    int32x4 z4 = {0, 0, 0, 0};                    // groups 2/3 unused (<=2D tensor)
    int32x8 z8 = {0, 0, 0, 0, 0, 0, 0, 0};
    __builtin_amdgcn_tensor_load_to_lds(g0, g1, z4, z4, z8, 0);
}

// ---------------------------------------------------------------------------
// Kernel 1: per-(b,h) prep.  Gathers 4 softsigned boundaries + both masks,
// packs as 2 x f32x4 per bh into ws:  ws[2*bh] = boundaries, ws[2*bh+1] = {cat,ord,0,0}
// ---------------------------------------------------------------------------
__global__ __launch_bounds__(256) void ca_prep_kernel(const float* __restrict__ x,
                                                      const int* __restrict__ ind,
                                                      const float* __restrict__ cat_u,
                                                      const float* __restrict__ ord_u,
                                                      f32x4* __restrict__ ws) {
    int bh = blockIdx.x * 256 + threadIdx.x;   // 0 .. BH-1 (grid sized exactly)
    int i0 = ind[0 * BH + bh];
    int i1 = ind[1 * BH + bh];
    int i2 = ind[2 * BH + bh];
    int i3 = ind[3 * BH + bh];
    f32x4 b;
    b.x = softsign_f(x[(size_t)i0 * BH + bh]);
    b.y = softsign_f(x[(size_t)i1 * BH + bh]);
    b.z = softsign_f(x[(size_t)i2 * BH + bh]);
    b.w = softsign_f(x[(size_t)i3 * BH + bh]);
    float cm = (cat_u[bh] < 0.1f) ? 1.0f : 0.0f;
    float om = ((ord_u[bh] < 0.7f) && (cm != 0.0f)) ? 1.0f : 0.0f;
    f32x4 m;
    m.x = cm; m.y = om; m.z = 0.0f; m.w = 0.0f;
    ws[2 * bh + 0] = b;
    ws[2 * bh + 1] = m;
}

// ---------------------------------------------------------------------------
// Kernel 2: HBM-bound stream.  Block = 256 threads, covers 1024 bh x S_TILE rows.
// Params staged via one TDM load into LDS, then hoisted to registers.
// ---------------------------------------------------------------------------
__device__ __forceinline__ float ca_elem(float v, f32x4 b, f32x4 m,
                                         float t0, float t1, float t2, float t3, float t4) {
    float xs  = v / (1.0f + __builtin_fabsf(v));
    int   cnt = (xs > b.x) + (xs > b.y) + (xs > b.z) + (xs > b.w);
    float val = (float)cnt - 2.5f;                       // num_classes/2 = 2.5
    float rem = (cnt == 0) ? t0 : (cnt == 1) ? t1 : (cnt == 2) ? t2
              : (cnt == 3) ? t3 : t4;
    float r = (m.x != 0.0f) ? val : xs;                  // categorical positions
    r       = (m.y != 0.0f) ? rem : r;                   // ordered&categorical positions
    return r;
}

__global__ __launch_bounds__(256) void ca_main_kernel(const float* __restrict__ x,
                                                      const int* __restrict__ perm,
                                                      const f32x4* __restrict__ ws,
                                                      float* __restrict__ out) {
    __shared__ f32x4 lds_par[2 * BH_PER_BLOCK];   // 32 KiB packed params

    const int t   = (int)threadIdx.x;
    const int bh0 = (int)blockIdx.x * BH_PER_BLOCK;
    const int s0  = (int)blockIdx.y * S_TILE;

    // Uniform remap table from perm (faithful ((val==k)*perm[k]).sum()).
    float tb0 = 0.f, tb1 = 0.f, tb2 = 0.f, tb3 = 0.f, tb4 = 0.f;
    {
        float p[NC];
        #pragma unroll
        for (int k = 0; k < NC; ++k) p[k] = (float)perm[k];   // uniform -> s_load
        #pragma unroll
        for (int k = 0; k < NC; ++k) {
            float fk = (float)k;
            tb0 += ((0.0f - 2.5f) == fk) ? p[k] : 0.0f;
            tb1 += ((1.0f - 2.5f) == fk) ? p[k] : 0.0f;
            tb2 += ((2.0f - 2.5f) == fk) ? p[k] : 0.0f;
            tb3 += ((3.0f - 2.5f) == fk) ? p[k] : 0.0f;
            tb4 += ((4.0f - 2.5f) == fk) ? p[k] : 0.0f;
        }
    }

    // One wave issues the TDM DMA of this block's packed params (32 KiB).
    if (__builtin_amdgcn_readfirstlane(t >> 5) == 0) {
        tdm_load_1d_f32((uint32_t)(uintptr_t)&lds_par[0],
                        (const void*)(ws + (size_t)bh0 * 2),
                        2u * BH_PER_BLOCK * 4u /* f32 elements = 8192 */);
    }
    __builtin_amdgcn_s_wait_tensorcnt(0);
    __syncthreads();

    // Hoist this thread's params (4 bh lanes) into registers.
    f32x4 pb[4], pm[4];
    #pragma unroll
    for (int j = 0; j < 4; ++j) {
        pb[j] = lds_par[t * 8 + 2 * j + 0];
        pm[j] = lds_par[t * 8 + 2 * j + 1];
    }

    const f32x4* __restrict__ xv = (const f32x4*)x;
    f32x4* __restrict__       ov = (f32x4*)out;
    const size_t col = (size_t)blockIdx.x * 256 + (size_t)t;   // f32x4 column

    #pragma unroll 2
    for (int s = s0; s < s0 + S_TILE; ++s) {
        size_t idx = (size_t)s * (BH / 4) + col;
        f32x4 v = __builtin_nontemporal_load(&xv[idx]);        // streamed once: NT
        f32x4 r;
        r.x = ca_elem(v.x, pb[0], pm[0], tb0, tb1, tb2, tb3, tb4);
        r.y = ca_elem(v.y, pb[1], pm[1], tb0, tb1, tb2, tb3, tb4);
        r.z = ca_elem(v.z, pb[2], pm[2], tb0, tb1, tb2, tb3, tb4);
        r.w = ca_elem(v.w, pb[3], pm[3], tb0, tb1, tb2, tb3, tb4);
        __builtin_nontemporal_store(r, &ov[idx]);              // streamed once: NT
    }
}

// ---------------------------------------------------------------------------
extern "C" void kernel_launch(void* const* d_in, const int* in_sizes, int n_in,
                              void* d_out, int out_size, void* d_ws, size_t ws_size,
                              hipStream_t stream) {
    const float* x     = (const float*)d_in[0];   // [S,B,H] f32
    const int*   ind   = (const int*)d_in[1];     // [4,B,H] i32
    const float* cat_u = (const float*)d_in[2];   // [B,H] f32
    const float* ord_u = (const float*)d_in[3];   // [B,H] f32
    const int*   perm  = (const int*)d_in[4];     // [5] i32
    float*       out   = (float*)d_out;
    f32x4*       ws    = (f32x4*)d_ws;            // 2*BH f32x4 = 512 KiB

    ca_prep_kernel<<<BH / 256, 256, 0, stream>>>(x, ind, cat_u, ord_u, ws);

    dim3 grid(BH / BH_PER_BLOCK, S_DIM / S_TILE, 1);   // (16, 128)
    ca_main_kernel<<<grid, 256, 0, stream>>>(x, perm, ws, out);
}